// MapCollisionLoss_23553600651768
// MI455X (gfx1250) — compile-verified
//
#include <hip/hip_runtime.h>
#include <hip/hip_bf16.h>

typedef __attribute__((ext_vector_type(2)))  float v2f;
typedef __attribute__((ext_vector_type(8)))  float v8f;

#define NUM_B  32
#define NUM_N  8
#define NUM_T  52
#define MAP_H  224
#define MAP_W  224
#define NP     100     // points per box (10x10)
#define NPAD   112     // padded to 7 WMMA tiles of 16
#define NTILE  7
#define WAVES  8       // 256-thread block = 8 wave32
#define BIGN2  1.0e30f // row penalty: excludes colliding / padded rows from min

// Select one of 4 wave-uniform-register masks by runtime index (avoids scratch).
__device__ __forceinline__ unsigned pick4(unsigned m0, unsigned m1, unsigned m2,
                                          unsigned m3, int k) {
  unsigned lo = (k & 1) ? m1 : m0;
  unsigned hi = (k & 1) ? m3 : m2;
  return (k & 2) ? hi : lo;
}

struct Pt { float ax, ay; bool coll; };

__device__ __forceinline__ Pt compute_pt(
    int p, float lw0, float lw1, float cy, float sy, float posx, float posy,
    float r00, float r01, float r02, float r10, float r11, float r12,
    const float* __restrict__ dmap, int b) {
  Pt o; o.ax = 0.0f; o.ay = 0.0f; o.coll = false;
  if (p < NP) {
    const int li = p / 10, wi = p % 10;
    const float lcl = -0.5f + (float)li * (1.0f / 9.0f);
    const float lcw = -0.5f + (float)wi * (1.0f / 9.0f);
    const float ptx = lcl * lw0;
    const float pty = lcw * lw1;
    // pts @ [[c, s], [-s, c]] + pos
    o.ax = ptx * cy - pty * sy + posx;
    o.ay = ptx * sy + pty * cy + posy;
    const float pxf = o.ax * r00 + o.ay * r01 + r02;
    const float pyf = o.ax * r10 + o.ay * r11 + r12;
    int px = (int)pxf;  // trunc toward zero matches astype(int32)
    int py = (int)pyf;
    px = min(max(px, 0), MAP_W - 1);
    py = min(max(py, 0), MAP_H - 1);
    const float drv = dmap[((long)b * MAP_H + py) * MAP_W + px];
    o.coll = (drv < 0.5f);  // offroad == 1  <=>  drivable == 0
  }
  return o;
}

__global__ __launch_bounds__(256) void MapCollisionLoss_kernel(
    const float* __restrict__ x, const float* __restrict__ dmap,
    const float* __restrict__ extent, const float* __restrict__ rfa,
    float* __restrict__ out) {
  __shared__ float sAx[WAVES][NPAD];
  __shared__ float sAy[WAVES][NPAD];
  __shared__ float sN2[WAVES][NPAD];  // penalized: BIGN2 replaces n2 for coll/pad rows
  __shared__ float wsum[WAVES];

  const int bn   = blockIdx.x;        // 0 .. B*N-1, maps directly to out index
  const int b    = bn / NUM_N;
  const int tid  = threadIdx.x;
  const int wave = tid >> 5;
  const int lane = tid & 31;
  const bool lo  = (lane < 16);

  const float lw0  = extent[b * 3 + 0];
  const float lw1  = extent[b * 3 + 1];
  const float diag = __builtin_amdgcn_sqrtf(lw0 * lw0 + lw1 * lw1);
  const float r00 = rfa[b * 9 + 0], r01 = rfa[b * 9 + 1], r02 = rfa[b * 9 + 2];
  const float r10 = rfa[b * 9 + 3], r11 = rfa[b * 9 + 4], r12 = rfa[b * 9 + 5];

  float accLane = 0.0f;  // per-lane partial of sum over (t, q)

  for (int t = wave; t < NUM_T; t += WAVES) {   // one wave per trajectory step
    const long m = (long)bn * NUM_T + t;
    const float posx = x[m * 6 + 0];
    const float posy = x[m * 6 + 1];
    const float yaw  = x[m * 6 + 3];
    const float sy = __sinf(yaw);
    const float cy = __cosf(yaw);

    // --- stage padded points into this wave's LDS slabs; ballot coll bits ---
    unsigned c0, c1, c2, c3;
    {
      Pt a0 = compute_pt(lane +  0, lw0, lw1, cy, sy, posx, posy,
                         r00, r01, r02, r10, r11, r12, dmap, b);
      sAx[wave][lane +  0] = a0.ax; sAy[wave][lane +  0] = a0.ay;
      sN2[wave][lane +  0] = a0.coll ? BIGN2 : (a0.ax * a0.ax + a0.ay * a0.ay);
      c0 = (unsigned)(__ballot(a0.coll) & 0xFFFFFFFFull);

      Pt a1 = compute_pt(lane + 32, lw0, lw1, cy, sy, posx, posy,
                         r00, r01, r02, r10, r11, r12, dmap, b);
      sAx[wave][lane + 32] = a1.ax; sAy[wave][lane + 32] = a1.ay;
      sN2[wave][lane + 32] = a1.coll ? BIGN2 : (a1.ax * a1.ax + a1.ay * a1.ay);
      c1 = (unsigned)(__ballot(a1.coll) & 0xFFFFFFFFull);

      Pt a2 = compute_pt(lane + 64, lw0, lw1, cy, sy, posx, posy,
                         r00, r01, r02, r10, r11, r12, dmap, b);
      sAx[wave][lane + 64] = a2.ax; sAy[wave][lane + 64] = a2.ay;
      sN2[wave][lane + 64] = a2.coll ? BIGN2 : (a2.ax * a2.ax + a2.ay * a2.ay);
      c2 = (unsigned)(__ballot(a2.coll) & 0xFFFFFFFFull);

      Pt a3 = compute_pt(lane + 96, lw0, lw1, cy, sy, posx, posy,
                         r00, r01, r02, r10, r11, r12, dmap, b);
      if (lane + 96 < NPAD) {  // p = 96..111; pads 100..111 get the penalty
        const bool pen = a3.coll || (lane + 96 >= NP);
        sAx[wave][lane + 96] = a3.ax; sAy[wave][lane + 96] = a3.ay;
        sN2[wave][lane + 96] = pen ? BIGN2 : (a3.ax * a3.ax + a3.ay * a3.ay);
      }
      c3 = (unsigned)(__ballot(a3.coll) & 0xFFFFFFFFull);
    }
    // DS ops are issued in-order per wave; drain them before cross-lane reads.
    asm volatile("s_wait_dscnt 0" ::: "memory");

    const int per_step = __popc(c0) + __popc(c1) + __popc(c2) + __popc(c3);
    if (per_step == 0 || per_step == NP) continue;  // wave-uniform: !overlap
    // per_step < NP guarantees >=1 non-penalized row => min stays finite.

    // --- preload all 7 A-fragments (rows) into registers ---
    // f32 16x4 A layout: lanes 0-15 hold K=0,1 (row M=lane); lanes 16-31 K=2,3.
    // Row p: (-2x_p, -2y_p, n2_p(+penalty), 1)
    v2f afrag[NTILE];
#pragma unroll
    for (int i = 0; i < NTILE; ++i) {
      const int pr = 16 * i + (lane & 15);
      const float px_ = sAx[wave][pr];
      const float py_ = sAy[wave][pr];
      const float pn2 = sN2[wave][pr];
      v2f a;
      a.x = lo ? (-2.0f * px_) : pn2;
      a.y = lo ? (-2.0f * py_) : 1.0f;
      afrag[i] = a;
    }

    // --- 7 column tiles: 7 WMMAs each, pure vector-min post-processing ---
    for (int j = 0; j < NTILE; ++j) {
      const int q = 16 * j + (lane & 15);
      const float qx = sAx[wave][q];
      const float qy = sAy[wave][q];
      const float qn2 = qx * qx + qy * qy;  // true norm (columns unpenalized)
      // B 4x16 layout: lanes 0-15 = rows K=0,1; lanes 16-31 = rows K=2,3.
      // Col q: (x_q, y_q, 1, n2_q)
      v2f bfrag;
      bfrag.x = lo ? qx : 1.0f;
      bfrag.y = lo ? qy : qn2;

      v8f macc = {BIGN2, BIGN2, BIGN2, BIGN2, BIGN2, BIGN2, BIGN2, BIGN2};
#pragma unroll
      for (int i = 0; i < NTILE; ++i) {
        v8f c = {};
        const v8f d2 = __builtin_amdgcn_wmma_f32_16x16x4_f32(
            false, afrag[i], false, bfrag, (short)0, c, false, false);
#pragma unroll
        for (int r = 0; r < 8; ++r) macc[r] = fminf(macc[r], d2[r]);
      }
      // tree-reduce the 8 row-slots held by this lane
      float m = fminf(fminf(fminf(macc[0], macc[1]), fminf(macc[2], macc[3])),
                      fminf(fminf(macc[4], macc[5]), fminf(macc[6], macc[7])));
      // rows M=0..7 live in lanes 0-15, M=8..15 in lanes 16-31: merge halves
      m = fminf(m, __shfl_xor(m, 16, 32));
      // min_p dist = sqrt(max(min_p d2, 0))  (sqrt monotonic; 0-case coincides)
      const float dist = __builtin_amdgcn_sqrtf(fmaxf(m, 0.0f));

      const int collq = (pick4(c0, c1, c2, c3, q >> 5) >> (q & 31)) & 1;
      const float loss = (collq && lo) ? (1.0f - dist / diag) : 0.0f;
      accLane += loss;
    }
  }

  // wave tree-reduce, then block combine
  float v = accLane;
#pragma unroll
  for (int off = 16; off >= 1; off >>= 1) v += __shfl_xor(v, off, 32);
  if (lane == 0) wsum[wave] = v;
  __syncthreads();
  if (tid == 0) {
    float tot = 0.0f;
#pragma unroll
    for (int w = 0; w < WAVES; ++w) tot += wsum[w];
    out[bn] = tot;
  }
}

extern "C" void kernel_launch(void* const* d_in, const int* in_sizes, int n_in,
                              void* d_out, int out_size, void* d_ws, size_t ws_size,
                              hipStream_t stream) {
  (void)in_sizes; (void)n_in; (void)out_size; (void)d_ws; (void)ws_size;
  const float* x      = (const float*)d_in[0];  // (B,N,T,6)
  const float* dmap   = (const float*)d_in[1];  // (B,H,W)
  const float* extent = (const float*)d_in[2];  // (B,3)
  const float* rfa    = (const float*)d_in[3];  // (B,3,3)
  float* out = (float*)d_out;                   // (B,N)

  dim3 grid(NUM_B * NUM_N);   // 256 blocks: one per (b,n)
  dim3 block(256);            // 8 wave32 per block, waves stride over T
  hipLaunchKernelGGL(MapCollisionLoss_kernel, grid, block, 0, stream,
                     x, dmap, extent, rfa, out);
}